// Inter_Correlation_Block_43361989820778
// MI455X (gfx1250) — compile-verified
//
#include <hip/hip_runtime.h>

#define N_NODES 20000
#define N_EDGES 320000
#define IN_F    168
#define HID     256

typedef float v2f __attribute__((ext_vector_type(2)));
typedef float v8f __attribute__((ext_vector_type(8)));

// ---------------------------------------------------------------------------
// deg[dst] += ew  (segment sum of edge weights at target nodes)
// ---------------------------------------------------------------------------
__global__ __launch_bounds__(256) void deg_kernel(const long long* __restrict__ dst,
                                                  const float* __restrict__ ew,
                                                  float* __restrict__ deg) {
    int e = blockIdx.x * 256 + threadIdx.x;
    if (e < N_EDGES) {
        unsafeAtomicAdd(&deg[(int)dst[e]], ew[e]);
    }
}

// dinv[i] = rsqrt(deg[i] + 1)
__global__ __launch_bounds__(256) void dinv_kernel(const float* __restrict__ deg,
                                                   float* __restrict__ dinv) {
    int i = blockIdx.x * 256 + threadIdx.x;
    if (i < N_NODES) {
        float d = deg[i] + 1.0f;
        dinv[i] = (d > 0.0f) ? rsqrtf(d) : 0.0f;
    }
}

// norm[e] = dinv[src]*ew*dinv[dst]
__global__ __launch_bounds__(256) void norm_kernel(const long long* __restrict__ src,
                                                   const long long* __restrict__ dst,
                                                   const float* __restrict__ ew,
                                                   const float* __restrict__ dinv,
                                                   float* __restrict__ norm) {
    int e = blockIdx.x * 256 + threadIdx.x;
    if (e < N_EDGES) {
        norm[e] = dinv[(int)src[e]] * ew[e] * dinv[(int)dst[e]];
    }
}

// ---------------------------------------------------------------------------
// Y[M x N] = X[M x K] * W[N x K]^T   via V_WMMA_F32_16X16X4_F32
// One wave computes one 16x16 tile of Y; 8 waves per 256-thread block.
// grid = (N/16/8, M/16).  M % 16 == 0, N % 128 == 0, K % 4 == 0 (no masking,
// EXEC stays all-ones as WMMA requires).
//
// A fragment (16x4, 2 VGPRs): lanes 0-15 -> K=k,k+1 ; lanes 16-31 -> K=k+2,k+3
// B fragment (4x16, 2 VGPRs): VGPR v, lanes 0-15 = row v, lanes 16-31 = row v+2
// Both reduce to the same per-lane addressing because Y = X * W^T.
// ---------------------------------------------------------------------------
__global__ __launch_bounds__(256) void wmma_gemm_xwt(const float* __restrict__ X,
                                                     const float* __restrict__ W,
                                                     float* __restrict__ Y,
                                                     int M, int N, int K) {
    const int wave  = threadIdx.x >> 5;
    const int lane  = threadIdx.x & 31;
    const int half  = lane >> 4;   // which K pair (A) / which row pair (B)
    const int l     = lane & 15;   // M index (A) / N index (B)

    const int mbase = blockIdx.y << 4;
    const int nbase = (blockIdx.x * 8 + wave) << 4;

    const float* __restrict__ arow = X + (size_t)(mbase + l) * K;
    const float* __restrict__ brow = W + (size_t)(nbase + l) * K;

    v8f acc = {};
    for (int k = 0; k < K; k += 4) {
        const int ka = k + (half << 1);
        v2f a = *(const v2f*)(arow + ka);
        v2f b = *(const v2f*)(brow + ka);
        // D = A x B + C, f32 in / f32 out (full precision, matches reference)
        acc = __builtin_amdgcn_wmma_f32_16x16x4_f32(
            /*neg_a=*/false, a, /*neg_b=*/false, b,
            /*c_mod=*/(short)0, acc, /*reuse_a=*/false, /*reuse_b=*/false);
    }

    // C/D layout: VGPR v -> row (v) for lanes 0-15, row (v+8) for lanes 16-31
    float* __restrict__ yptr = Y + (size_t)(mbase + half * 8) * N + nbase + l;
#pragma unroll
    for (int v = 0; v < 8; ++v) {
        yptr[(size_t)v * N] = acc[v];
    }
}

// ---------------------------------------------------------------------------
// agg[dst] += norm[e] * xw[src]   — one wave per edge, 8 f32 per lane,
// hardware global_atomic_add_f32 (all traffic stays in the 192MB L2).
// ---------------------------------------------------------------------------
__global__ __launch_bounds__(256) void scatter_kernel(const long long* __restrict__ src,
                                                      const long long* __restrict__ dst,
                                                      const float* __restrict__ norm,
                                                      const float* __restrict__ xw,
                                                      float* __restrict__ agg) {
    const int wave = threadIdx.x >> 5;
    const int lane = threadIdx.x & 31;
    const int e = blockIdx.x * 8 + wave;
    if (e >= N_EDGES) return;

    const int s = (int)src[e];
    const int d = (int)dst[e];
    const float nv = norm[e];

    const float* __restrict__ xr = xw + (size_t)s * HID;
    float* __restrict__ ar = agg + (size_t)d * HID;

#pragma unroll
    for (int j = 0; j < HID / 32; ++j) {
        const int f = lane + j * 32;
        unsafeAtomicAdd(&ar[f], nv * xr[f]);
    }
}

// out = relu(agg + dinv^2 * xw + bias)
__global__ __launch_bounds__(256) void combine_kernel(const float* __restrict__ agg,
                                                      const float* __restrict__ xw,
                                                      const float* __restrict__ dinv,
                                                      const float* __restrict__ bias,
                                                      float* __restrict__ out) {
    const int idx = blockIdx.x * 256 + threadIdx.x;
    if (idx < N_NODES * HID) {
        const int i = idx >> 8;   // / HID
        const int f = idx & 255;  // % HID
        const float di = dinv[i];
        const float v = agg[idx] + di * di * xw[idx] + bias[f];
        out[idx] = fmaxf(v, 0.0f);
    }
}

// ---------------------------------------------------------------------------
extern "C" void kernel_launch(void* const* d_in, const int* in_sizes, int n_in,
                              void* d_out, int out_size, void* d_ws, size_t ws_size,
                              hipStream_t stream) {
    (void)in_sizes; (void)n_in; (void)out_size; (void)ws_size;

    const float*     x   = (const float*)d_in[0];
    const long long* ei  = (const long long*)d_in[1];   // int64 [2, E]
    const float*     ew  = (const float*)d_in[2];
    const float*     W1  = (const float*)d_in[3];
    const float*     b1  = (const float*)d_in[4];
    const float*     W2  = (const float*)d_in[5];
    const float*     b2  = (const float*)d_in[6];
    float*           out = (float*)d_out;

    const long long* srcp = ei;
    const long long* dstp = ei + N_EDGES;

    // workspace layout (floats)
    float* ws   = (float*)d_ws;
    float* deg  = ws;                               // 20480 (padded)
    float* dinv = ws + 20480;                       // 20480
    float* norm = ws + 40960;                       // 320000
    float* xw   = ws + 40960 + N_EDGES;             // N*HID
    float* agg  = xw + (size_t)N_NODES * HID;       // N*HID
    float* h    = agg + (size_t)N_NODES * HID;      // N*HID

    const size_t featBytes = (size_t)N_NODES * HID * sizeof(float);

    // ---- normalization factors (shared by both layers) ----
    hipMemsetAsync(deg, 0, N_NODES * sizeof(float), stream);
    deg_kernel <<<(N_EDGES + 255) / 256, 256, 0, stream>>>(dstp, ew, deg);
    dinv_kernel<<<(N_NODES + 255) / 256, 256, 0, stream>>>(deg, dinv);
    norm_kernel<<<(N_EDGES + 255) / 256, 256, 0, stream>>>(srcp, dstp, ew, dinv, norm);

    const dim3 gemmGrid(HID / 16 / 8, N_NODES / 16);   // (2, 1250)
    const int  scatBlocks = N_EDGES / 8;               // 40000
    const int  combBlocks = (N_NODES * HID + 255) / 256;

    // ---- layer 1 ----
    wmma_gemm_xwt<<<gemmGrid, 256, 0, stream>>>(x, W1, xw, N_NODES, HID, IN_F);
    hipMemsetAsync(agg, 0, featBytes, stream);
    scatter_kernel<<<scatBlocks, 256, 0, stream>>>(srcp, dstp, norm, xw, agg);
    combine_kernel<<<combBlocks, 256, 0, stream>>>(agg, xw, dinv, b1, h);

    // ---- layer 2 ----
    wmma_gemm_xwt<<<gemmGrid, 256, 0, stream>>>(h, W2, xw, N_NODES, HID, HID);
    hipMemsetAsync(agg, 0, featBytes, stream);
    scatter_kernel<<<scatBlocks, 256, 0, stream>>>(srcp, dstp, norm, xw, agg);
    combine_kernel<<<combBlocks, 256, 0, stream>>>(agg, xw, dinv, b2, out);
}